// TransitionPrototype_88596585382464
// MI455X (gfx1250) — compile-verified
//
#include <hip/hip_runtime.h>

typedef __attribute__((ext_vector_type(2))) float v2f;
typedef __attribute__((ext_vector_type(4))) float v4f;
typedef __attribute__((ext_vector_type(8))) float v8f;

namespace {
constexpr int kH = 256;           // hidden size
constexpr int kP = 64;            // prototypes
constexpr int kWavesPerBlock = 8; // 256 threads
constexpr int kThreads = kWavesPerBlock * 32;
constexpr int kRowsPerWave = 16;  // M tile per wave
constexpr int kRowsPerBlock = kWavesPerBlock * kRowsPerWave; // 128
}

// D = A(16x4) x B(4x16) + C, f32 WMMA; one wave owns a 16x64 output tile.
__global__ __launch_bounds__(kThreads) void proto_sim_wmma(
    const float* __restrict__ X,
    const int*   __restrict__ transitions,
    const float* __restrict__ protos,
    float*       __restrict__ out)
{
    // 64 KB prototypes (XOR-swizzled columns: bank-conflict-free B reads) + norms
    __shared__ float sP[kP * kH];
    __shared__ float sP2[kP];

    const int tid  = threadIdx.x;
    const int lane = tid & 31;
    const int wave = tid >> 5;
    const int m    = lane & 15;   // row/col within 16-slice
    const int hi   = lane >> 4;   // half-wave (K sub-slice)

    // ---- stage prototypes global -> LDS (float4, swizzled) ----
    for (int q = tid; q < (kP * kH) / 4; q += kThreads) {
        const int row = q >> 6;            // 64 quads per 256-wide row
        const int k   = (q & 63) << 2;
        const v4f v = *(const v4f*)(protos + row * kH + k);
        *(v4f*)(&sP[row * kH + (k ^ ((row & 15) << 2))]) = v;
    }
    __syncthreads();

    // ---- prototype norms once per block (swizzle is a within-row bijection,
    //      so a linear read covers the row; deterministic order, no atomics) ----
    if (tid < kP) {
        const v4f* r = (const v4f*)(&sP[tid * kH]);
        v2f s2 = {0.f, 0.f};
        #pragma unroll 8
        for (int i = 0; i < kH / 4; ++i) {
            const v4f v = r[i];
            s2.x += v.x * v.x + v.y * v.y;
            s2.y += v.z * v.z + v.w * v.w;
        }
        sP2[tid] = s2.x + s2.y;
    }
    __syncthreads();

    const int row0 = (blockIdx.x * kWavesPerBlock + wave) * kRowsPerWave;
    const int koff = hi << 1;              // K offset for this half-wave: 0 or 2
    const int kswz = m << 2;               // per-row LDS swizzle

    const float* xrow = X + (size_t)(row0 + m) * kH + koff;

    v8f acc0 = {}, acc1 = {}, acc2 = {}, acc3 = {};
    v2f x2p = {0.f, 0.f};

    #pragma unroll 4
    for (int k = 0; k < kH; k += 4) {
        // A fragment: X[m, k+2*hi .. +1]  (ISA 32-bit A layout: VGPR0=K even, VGPR1=K odd)
        const v2f a = *(const v2f*)(xrow + k);

        const int kk = (k + koff) ^ kswz;  // swizzled column pair (contiguous, 8B aligned)
        const v2f b0 = *(const v2f*)(&sP[( 0 + m) * kH + kk]);
        const v2f b1 = *(const v2f*)(&sP[(16 + m) * kH + kk]);
        const v2f b2 = *(const v2f*)(&sP[(32 + m) * kH + kk]);
        const v2f b3 = *(const v2f*)(&sP[(48 + m) * kH + kk]);

        x2p = x2p + a * a;                 // single v_pk_fma_f32 per iteration

        acc0 = __builtin_amdgcn_wmma_f32_16x16x4_f32(false, a, false, b0, (short)0, acc0, false, false);
        acc1 = __builtin_amdgcn_wmma_f32_16x16x4_f32(false, a, false, b1, (short)0, acc1, false, false);
        acc2 = __builtin_amdgcn_wmma_f32_16x16x4_f32(false, a, false, b2, (short)0, acc2, false, false);
        acc3 = __builtin_amdgcn_wmma_f32_16x16x4_f32(false, a, false, b3, (short)0, acc3, false, false);
    }

    // horizontal + cross-half-wave combine of ||x||^2 (wave32: xor lane 16)
    float x2 = x2p.x + x2p.y;
    x2 += __shfl_xor(x2, 16, 32);

    // per-column prototype norms (lane L covers proto (L&15)+16*t, matching C layout)
    const float p2t0 = sP2[ 0 + m];
    const float p2t1 = sP2[16 + m];
    const float p2t2 = sP2[32 + m];
    const float p2t3 = sP2[48 + m];

    const int tr = transitions[row0 + m]; // lane L holds flag of row (L&15)

    // C/D layout: VGPR r -> row r + 8*hi, col = (lane&15) + 16*tile
    #pragma unroll
    for (int r = 0; r < 8; ++r) {
        const int src = r + (hi << 3);            // row within tile for this acc VGPR
        const float x2v = __shfl(x2, src, 32);    // broadcast row norm
        const int   trv = __shfl(tr, src, 32);    // broadcast row's transition flag
        const float mT = trv ? 1.f : 0.f;         // protos [0,16): transition mask
        const float mR = 1.f - mT;                // protos [16,64): regular mask

        float* orow = out + (size_t)(row0 + src) * kP;
        orow[ 0 + m] = __expf(-(x2v + p2t0 - 2.f * acc0[r])) * mT;
        orow[16 + m] = __expf(-(x2v + p2t1 - 2.f * acc1[r])) * mR;
        orow[32 + m] = __expf(-(x2v + p2t2 - 2.f * acc2[r])) * mR;
        orow[48 + m] = __expf(-(x2v + p2t3 - 2.f * acc3[r])) * mR;
    }
}

extern "C" void kernel_launch(void* const* d_in, const int* in_sizes, int n_in,
                              void* d_out, int out_size, void* d_ws, size_t ws_size,
                              hipStream_t stream) {
    const float* X           = (const float*)d_in[0];
    const int*   transitions = (const int*)d_in[1];
    const float* protos      = (const float*)d_in[2];
    float*       out         = (float*)d_out;

    const int B    = in_sizes[1];                             // 131072 rows
    const int grid = (B + kRowsPerBlock - 1) / kRowsPerBlock; // 1024 blocks

    proto_sim_wmma<<<grid, kThreads, 0, stream>>>(X, transitions, protos, out);
}